// Net_58729382805607
// MI455X (gfx1250) — compile-verified
//
#include <hip/hip_runtime.h>

typedef __attribute__((ext_vector_type(2))) float v2f;
typedef __attribute__((ext_vector_type(8))) float v8f;

#define F_IN  256
#define HID   64
#define C_OUT 40
#define C_PAD 48

// ---------------- degree + gnn_edge ----------------
__global__ __launch_bounds__(256) void k_deg_gnn(const int* __restrict__ dst,
                                                 const float* __restrict__ ew,
                                                 float* __restrict__ deg,
                                                 float* __restrict__ out_gnn,
                                                 int E) {
    int e = blockIdx.x * blockDim.x + threadIdx.x;
    if (e >= E) return;
    atomicAdd(&deg[dst[e]], 1.0f);
    out_gnn[e] = (ew[e] > 0.5f) ? 1.0f : -1.0f;
}

__global__ __launch_bounds__(256) void k_dinv(float* __restrict__ deg_dinv, int n) {
    int i = blockIdx.x * blockDim.x + threadIdx.x;
    if (i >= n) return;
    // deg includes self loop (+1); always > 0 so rsqrt branch of reference
    deg_dinv[i] = rsqrtf(deg_dinv[i] + 1.0f);
}

// ---------------- GEMM1: xw1[n,64] = x[n,256] @ W1[256,64] (f32 WMMA) ----------------
__global__ __launch_bounds__(256) void k_gemm1(const float* __restrict__ x,
                                               const float* __restrict__ W1,
                                               float* __restrict__ xw1, int n) {
    __shared__ float sB[F_IN * HID];  // 64 KB, fits 320KB/WGP LDS
    const int tid = threadIdx.x;
    {   // cooperative vectorized stage of W1 into LDS
        const float4* Wv = (const float4*)W1;
        float4* sv = (float4*)sB;
        #pragma unroll
        for (int i = tid; i < (F_IN * HID) / 4; i += 256) sv[i] = Wv[i];
    }
    __syncthreads();

    const int wave  = tid >> 5;
    const int lane  = tid & 31;
    const int mtile = wave >> 2;         // 0..1
    const int ntile = wave & 3;          // 0..3
    const int m0    = blockIdx.x * 32 + mtile * 16;
    const int n0    = ntile * 16;
    const int lhalf = lane & 15;
    const int khi   = (lane >> 4) * 2;   // lanes 16..31 hold K=k+2,k+3

    int arow = m0 + lhalf;
    if (arow >= n) arow = 0;             // clamp; stores guarded below
    const float* aptr = x + (size_t)arow * F_IN + khi;
    const float* bcol = sB + n0 + lhalf;

    v8f c = {};
    #pragma unroll 8
    for (int k = 0; k < F_IN; k += 4) {
        v2f a = *(const v2f*)(aptr + k);                 // A 16x4: 2 f32/lane
        v2f b;
        b.x = bcol[(k + khi) * HID];                     // B 4x16: 2 f32/lane
        b.y = bcol[(k + khi + 1) * HID];
        c = __builtin_amdgcn_wmma_f32_16x16x4_f32(false, a, false, b,
                                                  (short)0, c, false, false);
    }

    const int col   = n0 + lhalf;
    const int rbase = m0 + ((lane >> 4) << 3);           // lanes>=16 -> M+8
    #pragma unroll
    for (int v = 0; v < 8; ++v) {
        int r = rbase + v;
        if (r < n) xw1[(size_t)r * HID + col] = c[v];
    }
}

// ---------------- self-loop init: agg = xw * dinv[i]^2 ----------------
template <int F>
__global__ __launch_bounds__(256) void k_selfinit(const float* __restrict__ xw,
                                                  const float* __restrict__ dinv,
                                                  float* __restrict__ agg,
                                                  long long total) {
    long long gid = (long long)blockIdx.x * blockDim.x + threadIdx.x;
    if (gid >= total) return;
    int node = (int)(gid / F);
    float dv = dinv[node];
    agg[gid] = xw[gid] * dv * dv;
}

// ---------------- edge scatter-add: agg[d] += xw[s] * dinv[s]*dinv[d] ----------------
template <int F>
__global__ __launch_bounds__(256) void k_edge_agg(const int* __restrict__ src,
                                                  const int* __restrict__ dst,
                                                  const float* __restrict__ dinv,
                                                  const float* __restrict__ xw,
                                                  float* __restrict__ agg,
                                                  long long total) {
    long long gid = (long long)blockIdx.x * blockDim.x + threadIdx.x;
    if (gid >= total) return;
    int e = (int)(gid / F);
    int f = (int)(gid - (long long)e * F);
    int s = src[e], d = dst[e];
    float nrm = dinv[s] * dinv[d];
    atomicAdd(&agg[(size_t)d * F + f], xw[(size_t)s * F + f] * nrm);
}

// ---------------- bias + relu in place (layer 1) ----------------
__global__ __launch_bounds__(256) void k_bias_relu(float* __restrict__ a,
                                                   const float* __restrict__ b1,
                                                   long long total) {
    long long gid = (long long)blockIdx.x * blockDim.x + threadIdx.x;
    if (gid >= total) return;
    a[gid] = fmaxf(a[gid] + b1[gid & (HID - 1)], 0.0f);
}

// ---------------- GEMM2: xw2[n,40] = h[n,64] @ W2[64,40] (f32 WMMA, N padded to 48) ----
__global__ __launch_bounds__(96) void k_gemm2(const float* __restrict__ h,
                                              const float* __restrict__ W2,
                                              float* __restrict__ xw2, int n) {
    __shared__ float sB[HID * C_PAD];   // 12 KB, zero padded cols 40..47
    const int tid = threadIdx.x;
    for (int idx = tid; idx < HID * C_PAD; idx += 96) {
        int r = idx / C_PAD, cc = idx - r * C_PAD;
        sB[idx] = (cc < C_OUT) ? W2[r * C_OUT + cc] : 0.0f;
    }
    __syncthreads();

    const int wave  = tid >> 5;          // 3 waves -> 3 N tiles (48 cols)
    const int lane  = tid & 31;
    const int n0    = wave * 16;
    const int m0    = blockIdx.x * 16;
    const int lhalf = lane & 15;
    const int khi   = (lane >> 4) * 2;

    int arow = m0 + lhalf;
    if (arow >= n) arow = 0;
    const float* aptr = h + (size_t)arow * HID + khi;
    const float* bcol = sB + n0 + lhalf;

    v8f c = {};
    #pragma unroll
    for (int k = 0; k < HID; k += 4) {
        v2f a = *(const v2f*)(aptr + k);
        v2f b;
        b.x = bcol[(k + khi) * C_PAD];
        b.y = bcol[(k + khi + 1) * C_PAD];
        c = __builtin_amdgcn_wmma_f32_16x16x4_f32(false, a, false, b,
                                                  (short)0, c, false, false);
    }

    const int col   = n0 + lhalf;
    const int rbase = m0 + ((lane >> 4) << 3);
    if (col < C_OUT) {
        #pragma unroll
        for (int v = 0; v < 8; ++v) {
            int r = rbase + v;
            if (r < n) xw2[(size_t)r * C_OUT + col] = c[v];
        }
    }
}

// ---------------- per-node: +b2, x_out, log_softmax, softmax ----------------
__global__ __launch_bounds__(256) void k_softmax(const float* __restrict__ agg2,
                                                 const float* __restrict__ b2,
                                                 float* __restrict__ out_log,
                                                 float* __restrict__ out_x,
                                                 float* __restrict__ out_soft,
                                                 int n) {
    int i = blockIdx.x * blockDim.x + threadIdx.x;
    if (i >= n) return;
    float xc[C_OUT];
    const float4* rp = (const float4*)(agg2 + (size_t)i * C_OUT);
    const float4* bp = (const float4*)b2;
    float mx = -3.4e38f;
    #pragma unroll
    for (int q = 0; q < C_OUT / 4; ++q) {
        float4 v = rp[q], bb = bp[q];
        xc[4 * q + 0] = v.x + bb.x;
        xc[4 * q + 1] = v.y + bb.y;
        xc[4 * q + 2] = v.z + bb.z;
        xc[4 * q + 3] = v.w + bb.w;
        mx = fmaxf(mx, fmaxf(fmaxf(xc[4 * q], xc[4 * q + 1]),
                             fmaxf(xc[4 * q + 2], xc[4 * q + 3])));
    }
    float se = 0.0f;
    float ex[C_OUT];
    #pragma unroll
    for (int c = 0; c < C_OUT; ++c) { ex[c] = expf(xc[c] - mx); se += ex[c]; }
    float lse = logf(se);
    float inv = 1.0f / se;
    float4* po = (float4*)(out_x + (size_t)i * C_OUT);
    float4* pl = (float4*)(out_log + (size_t)i * C_OUT);
    float4* ps = (float4*)(out_soft + (size_t)i * C_OUT);
    #pragma unroll
    for (int q = 0; q < C_OUT / 4; ++q) {
        float4 vo, vl, vs;
        vo.x = xc[4 * q + 0]; vo.y = xc[4 * q + 1]; vo.z = xc[4 * q + 2]; vo.w = xc[4 * q + 3];
        vl.x = xc[4 * q + 0] - mx - lse; vl.y = xc[4 * q + 1] - mx - lse;
        vl.z = xc[4 * q + 2] - mx - lse; vl.w = xc[4 * q + 3] - mx - lse;
        vs.x = ex[4 * q + 0] * inv; vs.y = ex[4 * q + 1] * inv;
        vs.z = ex[4 * q + 2] * inv; vs.w = ex[4 * q + 3] * inv;
        po[q] = vo; pl[q] = vl; ps[q] = vs;
    }
}

// ---------------- per-edge cosine dissimilarity ----------------
__global__ __launch_bounds__(256) void k_cosine(const int* __restrict__ src,
                                                const int* __restrict__ dst,
                                                const float* __restrict__ xo,
                                                float* __restrict__ out_cos,
                                                int E) {
    int e = blockIdx.x * blockDim.x + threadIdx.x;
    if (e >= E) return;
    const float4* a = (const float4*)(xo + (size_t)src[e] * C_OUT);
    const float4* b = (const float4*)(xo + (size_t)dst[e] * C_OUT);
    float dot = 0.0f, na = 0.0f, nb = 0.0f;
    #pragma unroll
    for (int q = 0; q < C_OUT / 4; ++q) {
        float4 av = a[q], bv = b[q];
        dot += av.x * bv.x + av.y * bv.y + av.z * bv.z + av.w * bv.w;
        na  += av.x * av.x + av.y * av.y + av.z * av.z + av.w * av.w;
        nb  += bv.x * bv.x + bv.y * bv.y + bv.z * bv.z + bv.w * bv.w;
    }
    float den = fmaxf(sqrtf(na), 1e-8f) * fmaxf(sqrtf(nb), 1e-8f);
    out_cos[e] = 1.0f - dot / den;
}

extern "C" void kernel_launch(void* const* d_in, const int* in_sizes, int n_in,
                              void* d_out, int out_size, void* d_ws, size_t ws_size,
                              hipStream_t stream) {
    const float* x  = (const float*)d_in[0];
    const int*   ei = (const int*)d_in[1];
    const float* ew = (const float*)d_in[2];
    const float* W1 = (const float*)d_in[3];
    const float* b1 = (const float*)d_in[4];
    const float* W2 = (const float*)d_in[5];
    const float* b2 = (const float*)d_in[6];

    const int n = in_sizes[0] / F_IN;
    const int E = in_sizes[2];
    const int* src = ei;
    const int* dst = ei + E;

    // workspace layout (floats): dinv | xw1(n*64, reused as xw2) | agg1(n*64) | agg2(n*40)
    float* ws   = (float*)d_ws;
    float* dinv = ws;
    float* xw1  = dinv + n;
    float* agg1 = xw1 + (size_t)n * HID;
    float* agg2 = agg1 + (size_t)n * HID;
    float* xw2  = xw1;   // gemm1 output consumed before gemm2 writes here

    // output layout: log_softmax | x_out | cosine | gnn_edge | softmax
    float* out      = (float*)d_out;
    size_t NC       = (size_t)n * C_OUT;
    float* out_log  = out;
    float* out_x    = out + NC;
    float* out_cos  = out + 2 * NC;
    float* out_gnn  = out + 2 * NC + E;
    float* out_soft = out + 2 * NC + 2 * (size_t)E;

    const long long tot1 = (long long)n * HID;
    const long long totE1 = (long long)E * HID;
    const long long tot2 = (long long)n * C_OUT;
    const long long totE2 = (long long)E * C_OUT;

    hipMemsetAsync(dinv, 0, (size_t)n * sizeof(float), stream);
    k_deg_gnn<<<(E + 255) / 256, 256, 0, stream>>>(dst, ew, dinv, out_gnn, E);
    k_dinv<<<(n + 255) / 256, 256, 0, stream>>>(dinv, n);

    // layer 1
    k_gemm1<<<(n + 31) / 32, 256, 0, stream>>>(x, W1, xw1, n);
    k_selfinit<HID><<<(int)((tot1 + 255) / 256), 256, 0, stream>>>(xw1, dinv, agg1, tot1);
    k_edge_agg<HID><<<(int)((totE1 + 255) / 256), 256, 0, stream>>>(src, dst, dinv, xw1, agg1, totE1);
    k_bias_relu<<<(int)((tot1 + 255) / 256), 256, 0, stream>>>(agg1, b1, tot1);

    // layer 2
    k_gemm2<<<(n + 15) / 16, 96, 0, stream>>>(agg1, W2, xw2, n);
    k_selfinit<C_OUT><<<(int)((tot2 + 255) / 256), 256, 0, stream>>>(xw2, dinv, agg2, tot2);
    k_edge_agg<C_OUT><<<(int)((totE2 + 255) / 256), 256, 0, stream>>>(src, dst, dinv, xw2, agg2, totE2);

    // heads
    k_softmax<<<(n + 255) / 256, 256, 0, stream>>>(agg2, b2, out_log, out_x, out_soft, n);
    k_cosine<<<(E + 255) / 256, 256, 0, stream>>>(src, dst, out_x, out_cos, E);
}